// EGNN_40424232190561
// MI455X (gfx1250) — compile-verified
//
#include <hip/hip_runtime.h>

// ---------------- problem constants (from reference) ----------------
constexpr int   kN    = 100000;
constexpr int   kE    = 1600000;
constexpr int   kFin  = 128;
constexpr int   kH    = 64;
constexpr int   kL    = 4;
constexpr int   kC    = 40;
constexpr float kAlpha = 0.1f;                   // ALPHA
constexpr float kResW  = 0.2f - 0.1f;            // c_min - alpha = 0.1
constexpr float kAggW  = 1.0f - kResW - kAlpha;  // 0.8

typedef float v2f __attribute__((ext_vector_type(2)));
typedef float v8f __attribute__((ext_vector_type(8)));

// fp32 WMMA 16x16x4 (CDNA5): D = A(16x4) * B(4x16) + C(16x16), full fp32.
static __device__ __forceinline__ v8f wmma_f32(v2f a, v2f b, v8f c) {
  return __builtin_amdgcn_wmma_f32_16x16x4_f32(
      /*neg_a=*/false, a, /*neg_b=*/false, b,
      /*c_mod=*/(short)0, c, /*reuse_a=*/false, /*reuse_b=*/false);
}

static __device__ __forceinline__ v2f f2v(float2 f) { v2f r; r.x = f.x; r.y = f.y; return r; }

// Hardware FP32 atomic add (no CAS loop); untracked store -> S_ENDPGM's
// implicit wait-idle covers completion.
static __device__ __forceinline__ void atomic_add_f32(float* p, float v) {
  asm volatile("global_atomic_add_f32 %0, %1, off" :: "v"(p), "v"(v) : "memory");
}

// Async global -> LDS copy (16B per lane), tracked by ASYNCcnt.
// lds_off is the wave-relative LDS byte offset (low 32 bits of a generic
// pointer to a __shared__ object are the LDS offset on CDNA5).
static __device__ __forceinline__ void async_copy_b128(unsigned lds_off, const void* g) {
  asm volatile("global_load_async_to_lds_b128 %0, %1, off"
               :: "v"(lds_off), "v"(g) : "memory");
}
static __device__ __forceinline__ void wait_asynccnt0() {
  asm volatile("s_wait_asynccnt 0x0" ::: "memory");
}
static __device__ __forceinline__ unsigned lds_off_u32(const void* p) {
  return (unsigned)(unsigned long long)p;
}

// ---------------- utility kernels ----------------
__global__ void __launch_bounds__(256) egnn_zero(float* __restrict__ p, int n) {
  int i = blockIdx.x * 256 + threadIdx.x;
  int stride = gridDim.x * 256;
  for (; i < n; i += stride) p[i] = 0.0f;
}

// deg[col] += 1 over all E edges (self-loop +1 folded into egnn_dis)
__global__ void __launch_bounds__(256) egnn_deg(const long long* __restrict__ ei,
                                                float* __restrict__ deg) {
  int e = blockIdx.x * 256 + threadIdx.x;
  if (e >= kE) return;
  long long c = ei[(size_t)kE + e];
  atomic_add_f32(deg + c, 1.0f);
}

// dis[i] = (deg[i] + 1)^-0.5   (deg+1 >= 1 always, matches reference guard)
__global__ void __launch_bounds__(256) egnn_dis(float* __restrict__ d) {
  int i = blockIdx.x * 256 + threadIdx.x;
  if (i < kN) d[i] = rsqrtf(d[i] + 1.0f);
}

// ---------------- input GEMM: x0 = xa = relu(x @ W_in + b_in) ----------------
// Pair-row stride (in float2) for H=64: 80 float2 = 640B ==> half-waves hit
// disjoint LDS bank halves (640B = 160 DWORDs == 32 mod 64).
constexpr int kPJ64 = 80;

__global__ void __launch_bounds__(256) egnn_gemm_in(
    const float* __restrict__ x, const float* __restrict__ Win,
    const float* __restrict__ bin, float* __restrict__ x0, float* __restrict__ xa)
{
  __shared__ float  wraw[kFin * kH];              // 32 KB raw staging
  __shared__ float2 wp[(kFin / 2) * kPJ64];       // 40 KB pair-interleaved
  {
    // async stage raw W_in into LDS (2048 x 16B chunks)
    unsigned lbase = lds_off_u32(wraw);
    const char* g = (const char*)Win;
    for (int i = threadIdx.x; i < (kFin * kH * 4) / 16; i += 256)
      async_copy_b128(lbase + i * 16, g + i * 16);
    wait_asynccnt0();
  }
  __syncthreads();
  // swizzle to pair layout: wp[kp*PJ + n] = (W[2kp][n], W[2kp+1][n])
  for (int i = threadIdx.x; i < (kFin / 2) * kH; i += 256) {
    int kp = i / kH, n = i % kH;
    wp[kp * kPJ64 + n] = make_float2(wraw[(2 * kp) * kH + n],
                                     wraw[(2 * kp + 1) * kH + n]);
  }
  __syncthreads();

  const int wave = threadIdx.x >> 5;
  const int tile = blockIdx.x * 8 + wave;      // 16-row M tile
  if (tile >= kN / 16) return;                 // wave-uniform -> EXEC all-ones
  const int lane = threadIdx.x & 31;
  const int lr   = lane & 15;
  const int half = lane >> 4;

  const float2* xr = (const float2*)(x + (size_t)(tile * 16 + lr) * kFin);
  v8f c0 = {}, c1 = {}, c2 = {}, c3 = {};

#pragma unroll 4
  for (int k0 = 0; k0 < kFin; k0 += 4) {
    float2 av = xr[(k0 >> 1) + half];          // A: K(k0|k0+2), K(k0+1|k0+3)
    v2f a = f2v(av);
    const float2* bp = wp + ((k0 >> 1) + half) * kPJ64;
    c0 = wmma_f32(a, f2v(bp[lr]),      c0);
    c1 = wmma_f32(a, f2v(bp[lr + 16]), c1);
    c2 = wmma_f32(a, f2v(bp[lr + 32]), c2);
    c3 = wmma_f32(a, f2v(bp[lr + 48]), c3);
  }

  const int mbase = tile * 16 + half * 8;
  const float bb0 = bin[lr], bb1 = bin[lr + 16], bb2 = bin[lr + 32], bb3 = bin[lr + 48];
#pragma unroll
  for (int r = 0; r < 8; ++r) {
    size_t o = (size_t)(mbase + r) * kH;
    float v;
    v = fmaxf(c0[r] + bb0, 0.0f); x0[o + lr]      = v; xa[o + lr]      = v;
    v = fmaxf(c1[r] + bb1, 0.0f); x0[o + lr + 16] = v; xa[o + lr + 16] = v;
    v = fmaxf(c2[r] + bb2, 0.0f); x0[o + lr + 32] = v; xa[o + lr + 32] = v;
    v = fmaxf(c3[r] + bb3, 0.0f); x0[o + lr + 48] = v; xa[o + lr + 48] = v;
  }
}

// ---------------- edge scatter: agg[col] += norm * x[row] ----------------
// 16 threads/edge, float4 per thread; atomics land in L2 (agg = 25.6MB << 192MB).
__global__ void __launch_bounds__(256) egnn_scatter(
    const long long* __restrict__ ei, const float* __restrict__ dis,
    const float* __restrict__ xin, float* __restrict__ agg)
{
  const long long tid = (long long)blockIdx.x * 256 + threadIdx.x;
  const int e = (int)(tid >> 4);
  const int f = ((int)tid & 15) << 2;
  if (e >= kE) return;
  const long long r = ei[e];
  const long long c = ei[(size_t)kE + e];
  const float wgt = dis[r] * dis[c];
  const float4 v = *(const float4*)(xin + (size_t)r * kH + f);
  float* dst = agg + (size_t)c * kH + f;
  atomic_add_f32(dst + 0, v.x * wgt);
  atomic_add_f32(dst + 1, v.y * wgt);
  atomic_add_f32(dst + 2, v.z * wgt);
  atomic_add_f32(dst + 3, v.w * wgt);
}

// ---------------- fused layer: h = mix(agg,x,x0); x' = srelu(h @ Wg) ----------------
// Self-loop folded: h = AggW*agg + (AggW*d^2 + ResW)*x + Alpha*x0
__global__ void __launch_bounds__(256) egnn_layer(
    const float* __restrict__ agg, const float* __restrict__ xin,
    const float* __restrict__ x0, const float* __restrict__ dis,
    const float* __restrict__ Wg, const float* __restrict__ sb,
    float* __restrict__ xout)
{
  __shared__ float  wraw[kH * kH];            // 16 KB raw staging
  __shared__ float2 wp[(kH / 2) * kPJ64];     // 20 KB pair-interleaved
  {
    unsigned lbase = lds_off_u32(wraw);
    const char* g = (const char*)Wg;
    for (int i = threadIdx.x; i < (kH * kH * 4) / 16; i += 256)
      async_copy_b128(lbase + i * 16, g + i * 16);
    wait_asynccnt0();
  }
  __syncthreads();
  for (int i = threadIdx.x; i < (kH / 2) * kH; i += 256) {
    int kp = i / kH, n = i % kH;
    wp[kp * kPJ64 + n] = make_float2(wraw[(2 * kp) * kH + n],
                                     wraw[(2 * kp + 1) * kH + n]);
  }
  __syncthreads();

  const int wave = threadIdx.x >> 5;
  const int tile = blockIdx.x * 8 + wave;
  if (tile >= kN / 16) return;
  const int lane = threadIdx.x & 31;
  const int lr   = lane & 15;
  const int half = lane >> 4;

  const int m = tile * 16 + lr;
  const float d    = dis[m];
  const float coef = kAggW * d * d + kResW;
  const float2* ag  = (const float2*)(agg + (size_t)m * kH);
  const float2* xv  = (const float2*)(xin + (size_t)m * kH);
  const float2* x0v = (const float2*)(x0  + (size_t)m * kH);

  v8f c0 = {}, c1 = {}, c2 = {}, c3 = {};
#pragma unroll 4
  for (int k0 = 0; k0 < kH; k0 += 4) {
    const int idx = (k0 >> 1) + half;
    float2 A = ag[idx], X = xv[idx], X0 = x0v[idx];
    v2f a;
    a.x = kAggW * A.x + coef * X.x + kAlpha * X0.x;
    a.y = kAggW * A.y + coef * X.y + kAlpha * X0.y;
    const float2* bp = wp + idx * kPJ64;
    c0 = wmma_f32(a, f2v(bp[lr]),      c0);
    c1 = wmma_f32(a, f2v(bp[lr + 16]), c1);
    c2 = wmma_f32(a, f2v(bp[lr + 32]), c2);
    c3 = wmma_f32(a, f2v(bp[lr + 48]), c3);
  }

  const int mbase = tile * 16 + half * 8;
  const float s0 = sb[lr], s1 = sb[lr + 16], s2 = sb[lr + 32], s3 = sb[lr + 48];
#pragma unroll
  for (int r = 0; r < 8; ++r) {
    size_t o = (size_t)(mbase + r) * kH;
    xout[o + lr]      = fmaxf(c0[r] - s0, 0.0f) + s0;  // SReLU
    xout[o + lr + 16] = fmaxf(c1[r] - s1, 0.0f) + s1;
    xout[o + lr + 32] = fmaxf(c2[r] - s2, 0.0f) + s2;
    xout[o + lr + 48] = fmaxf(c3[r] - s3, 0.0f) + s3;
  }
}

// ---------------- output GEMM: out = x @ W_out + b_out (C=40, pad to 48) ----------------
// Pair-row stride 48 float2 = 384B = 96 DWORDs == 32 mod 64: bank-friendly as-is.
constexpr int kPJ48 = 48;

__global__ void __launch_bounds__(256) egnn_out(
    const float* __restrict__ xin, const float* __restrict__ Wo,
    const float* __restrict__ bo, float* __restrict__ out)
{
  __shared__ float  wraw[kH * kC];            // 10 KB raw staging (64x40)
  __shared__ float2 wp[(kH / 2) * kPJ48];     // 12 KB pair-interleaved, zero-padded
  {
    unsigned lbase = lds_off_u32(wraw);
    const char* g = (const char*)Wo;
    for (int i = threadIdx.x; i < (kH * kC * 4) / 16; i += 256)  // 640 chunks
      async_copy_b128(lbase + i * 16, g + i * 16);
    wait_asynccnt0();
  }
  __syncthreads();
  for (int i = threadIdx.x; i < (kH / 2) * kPJ48; i += 256) {
    int kp = i / kPJ48, n = i % kPJ48;
    float2 v = make_float2(0.0f, 0.0f);
    if (n < kC) v = make_float2(wraw[(2 * kp) * kC + n], wraw[(2 * kp + 1) * kC + n]);
    wp[kp * kPJ48 + n] = v;
  }
  __syncthreads();

  const int wave = threadIdx.x >> 5;
  const int tile = blockIdx.x * 8 + wave;
  if (tile >= kN / 16) return;
  const int lane = threadIdx.x & 31;
  const int lr   = lane & 15;
  const int half = lane >> 4;

  const float2* xr = (const float2*)(xin + (size_t)(tile * 16 + lr) * kH);
  v8f c0 = {}, c1 = {}, c2 = {};
#pragma unroll 4
  for (int k0 = 0; k0 < kH; k0 += 4) {
    float2 av = xr[(k0 >> 1) + half];
    v2f a = f2v(av);
    const float2* bp = wp + ((k0 >> 1) + half) * kPJ48;
    c0 = wmma_f32(a, f2v(bp[lr]),      c0);
    c1 = wmma_f32(a, f2v(bp[lr + 16]), c1);
    c2 = wmma_f32(a, f2v(bp[lr + 32]), c2);
  }

  const int mbase = tile * 16 + half * 8;
  const float bb0 = bo[lr], bb1 = bo[lr + 16];
  const float bb2 = (lr + 32 < kC) ? bo[lr + 32] : 0.0f;
#pragma unroll
  for (int r = 0; r < 8; ++r) {
    size_t o = (size_t)(mbase + r) * kC;
    out[o + lr]      = c0[r] + bb0;
    out[o + lr + 16] = c1[r] + bb1;
    if (lr + 32 < kC) out[o + lr + 32] = c2[r] + bb2;
  }
}

// ---------------- host orchestration ----------------
extern "C" void kernel_launch(void* const* d_in, const int* in_sizes, int n_in,
                              void* d_out, int out_size, void* d_ws, size_t ws_size,
                              hipStream_t stream) {
  const float*     x   = (const float*)d_in[0];
  const long long* ei  = (const long long*)d_in[1];  // int64 per reference
  const float*     Win = (const float*)d_in[2];
  const float*     bin = (const float*)d_in[3];
  const float*     Wg  = (const float*)d_in[4];      // [L,H,H]
  const float*     sb  = (const float*)d_in[5];      // [L,H]
  const float*     Wo  = (const float*)d_in[6];
  const float*     bo  = (const float*)d_in[7];
  float* out = (float*)d_out;

  // workspace layout (floats): dis | x0 | xa | xb | agg  (~103 MB total)
  float* ws  = (float*)d_ws;
  float* dis = ws;
  float* x0  = dis + kN;                    // byte offset 400000 (16B aligned)
  float* xa  = x0 + (size_t)kN * kH;
  float* xb  = xa + (size_t)kN * kH;
  float* agg = xb + (size_t)kN * kH;

  const int tiles   = kN / 16;              // 6250 (exact)
  const int gblocks = (tiles + 7) / 8;      // 8 waves (M-tiles) per 256-thread block

  egnn_zero<<<2048, 256, 0, stream>>>(dis, kN);
  egnn_deg <<<kE / 256, 256, 0, stream>>>(ei, dis);
  egnn_dis <<<(kN + 255) / 256, 256, 0, stream>>>(dis);
  egnn_gemm_in<<<gblocks, 256, 0, stream>>>(x, Win, bin, x0, xa);

  float* cur = xa; float* nxt = xb;
  for (int l = 0; l < kL; ++l) {
    egnn_zero   <<<4096, 256, 0, stream>>>(agg, kN * kH);
    egnn_scatter<<<(kE * 16) / 256, 256, 0, stream>>>(ei, dis, cur, agg);
    egnn_layer  <<<gblocks, 256, 0, stream>>>(agg, cur, x0, dis,
        Wg + (size_t)l * kH * kH, sb + (size_t)l * kH, nxt);
    float* t = cur; cur = nxt; nxt = t;
  }
  egnn_out<<<gblocks, 256, 0, stream>>>(cur, Wo, bo, out);
}